// SinkhornSolver_46127948759547
// MI455X (gfx1250) — compile-verified
//
#include <hip/hip_runtime.h>
#include <hip/hip_bf16.h>

#define EPS_INV 10.0f          // 1/EPSILON, EPSILON = 0.1
#define B_ 8
#define S_ 2048
#define T_ 2048
#define ITERS 20
#define SCH 16                  // S chunks for column update
#define TCOLS 256               // columns per block in column update
#define RWAVES 4                // waves per row-update block (T split 4-way)

typedef __attribute__((ext_vector_type(2))) float v2f;
typedef __attribute__((ext_vector_type(8))) float v8f;

// ---------------------------------------------------------------------------
// init: log_mu = log(max(mu,tiny)), log_nu = log(max(nu,tiny)), u=v=0
// ---------------------------------------------------------------------------
__global__ void sinkhorn_init(const float* __restrict__ mu,
                              const float* __restrict__ nu,
                              float* __restrict__ log_mu,
                              float* __restrict__ log_nu,
                              float* __restrict__ log_u,
                              float* __restrict__ log_v) {
  int i = blockIdx.x * blockDim.x + threadIdx.x;
  if (i < B_ * S_) {
    log_mu[i] = __logf(fmaxf(mu[i], __FLT_MIN__));
    log_u[i] = 0.0f;
  }
  if (i < B_ * T_) {
    log_nu[i] = __logf(fmaxf(nu[i], __FLT_MIN__));
    log_v[i] = 0.0f;
  }
}

// ---------------------------------------------------------------------------
// row update: log_u[b,s] = log_mu[b,s] - LSE_t( log_v[b,t] - cost[b,s,t]/eps )
// One block = 4 waves over the SAME 16 rows, each wave owning T/4 columns.
// Pass 1: per-wave per-row max (A-layout lanes, shfl_xor(16) pair combine),
//         cross-wave max via LDS.
// Pass 2: each wave accumulates partial row sums of exp(x - globalmax) on the
//         matrix pipe via V_WMMA_F32_16X16X4_F32 (B = ones), partials summed
//         through LDS (exact: all partials share the global rowmax).
// ---------------------------------------------------------------------------
__global__ __launch_bounds__(32 * RWAVES) void sinkhorn_row_update(
    const float* __restrict__ cost,
    const float* __restrict__ log_mu,
    const float* __restrict__ log_v,
    float* __restrict__ log_u) {
  __shared__ float smax[RWAVES * 16];
  __shared__ float ssum[RWAVES * 16];

  const int tid  = threadIdx.x;
  const int lane = tid & 31;
  const int w    = tid >> 5;           // wave id 0..RWAVES-1
  const int blk  = blockIdx.x;
  const int b    = blk / (S_ / 16);
  const int s0   = (blk % (S_ / 16)) * 16;

  const int r    = lane & 15;          // row within tile (A-matrix M index)
  const int koff = (lane >> 4) * 2;    // K offset within each K=4 chunk

  const int tBeg = w * (T_ / RWAVES);
  const int tEnd = tBeg + (T_ / RWAVES);

  const float* __restrict__ crow = cost + ((size_t)(b * S_ + s0 + r)) * T_ + koff;
  const float* __restrict__ lvp  = log_v + (size_t)b * T_ + koff;

  // ---- pass 1: per-lane max over this wave's span
  float m = -__FLT_MAX__;
  for (int t = tBeg; t < tEnd; t += 4) {
    v2f c  = *(const v2f*)(crow + t);
    v2f lv = *(const v2f*)(lvp + t);
    float x0 = fmaf(-EPS_INV, c.x, lv.x);
    float x1 = fmaf(-EPS_INV, c.y, lv.y);
    m = fmaxf(m, fmaxf(x0, x1));
  }
  m = fmaxf(m, __shfl_xor(m, 16));     // wave-local rowmax of row (lane&15)
  if (lane < 16) smax[w * 16 + lane] = m;
  __syncthreads();
  // cross-wave global rowmax for this lane's row
#pragma unroll
  for (int ww = 0; ww < RWAVES; ++ww) m = fmaxf(m, smax[ww * 16 + r]);
  __syncthreads();

  // ---- pass 2: WMMA-accumulated partial row sums of exp(x - globalmax)
  v8f acc = {0.f, 0.f, 0.f, 0.f, 0.f, 0.f, 0.f, 0.f};
  v2f onesv; onesv.x = 1.0f; onesv.y = 1.0f;   // B = ones(4x16)
  for (int t = tBeg; t < tEnd; t += 4) {
    v2f c  = *(const v2f*)(crow + t);
    v2f lv = *(const v2f*)(lvp + t);
    v2f a;
    a.x = __expf(fmaf(-EPS_INV, c.x, lv.x) - m);
    a.y = __expf(fmaf(-EPS_INV, c.y, lv.y) - m);
    // D = A(16x4) * ones(4x16) + C  -> every column of D holds the row sums
    acc = __builtin_amdgcn_wmma_f32_16x16x4_f32(
        /*neg_a=*/false, a, /*neg_b=*/false, onesv,
        /*c_mod=*/(short)0, acc, /*reuse_a=*/false, /*reuse_b=*/false);
  }

  // D layout: VGPR j, lanes 0-15 -> row j; lanes 16-31 -> row 8+j.
#pragma unroll
  for (int j = 0; j < 8; ++j) {
    if (lane == 0)  ssum[w * 16 + j]     = acc[j];
    if (lane == 16) ssum[w * 16 + 8 + j] = acc[j];
  }
  __syncthreads();

  // combine partial sums; thread tid<16 owns row tid (its own m is that
  // row's global max since lane==row for lanes 0..15 of wave 0)
  if (tid < 16) {
    float total = 0.0f;
#pragma unroll
    for (int ww = 0; ww < RWAVES; ++ww) total += ssum[ww * 16 + tid];
    size_t o = (size_t)b * S_ + s0 + tid;
    log_u[o] = log_mu[o] - (m + __logf(total));
  }
}

// ---------------------------------------------------------------------------
// column update, stage 1: partial (max, sumexp) over an S-chunk.
// Thread j owns column t0+j -> fully coalesced row-major streaming of cost.
// ---------------------------------------------------------------------------
__global__ __launch_bounds__(TCOLS) void sinkhorn_col_partial(
    const float* __restrict__ cost,
    const float* __restrict__ log_u,
    float* __restrict__ pm,
    float* __restrict__ ps) {
  const int blk   = blockIdx.x;
  const int tile  = blk % (T_ / TCOLS);
  const int chunk = (blk / (T_ / TCOLS)) % SCH;
  const int b     = blk / ((T_ / TCOLS) * SCH);
  const int t     = tile * TCOLS + threadIdx.x;
  const int sBeg  = chunk * (S_ / SCH);
  const int sEnd  = sBeg + (S_ / SCH);

  const float* __restrict__ cp  = cost + ((size_t)b * S_) * T_ + t;
  const float* __restrict__ lup = log_u + (size_t)b * S_;

  float m = -__FLT_MAX__;
  for (int s = sBeg; s < sEnd; ++s) {
    float x = fmaf(-EPS_INV, cp[(size_t)s * T_], lup[s]);
    m = fmaxf(m, x);
  }
  float sum = 0.0f;
  for (int s = sBeg; s < sEnd; ++s) {
    float x = fmaf(-EPS_INV, cp[(size_t)s * T_], lup[s]);
    sum += __expf(x - m);
  }
  size_t idx = ((size_t)(b * SCH + chunk)) * T_ + t;
  pm[idx] = m;
  ps[idx] = sum;
}

// ---------------------------------------------------------------------------
// column update, stage 2: merge SCH partial logsumexps.
// ---------------------------------------------------------------------------
__global__ void sinkhorn_col_combine(const float* __restrict__ pm,
                                     const float* __restrict__ ps,
                                     const float* __restrict__ log_nu,
                                     float* __restrict__ log_v) {
  int i = blockIdx.x * blockDim.x + threadIdx.x;
  if (i >= B_ * T_) return;
  int b = i / T_;
  int t = i % T_;
  const float* __restrict__ pmp = pm + (size_t)b * SCH * T_ + t;
  const float* __restrict__ psp = ps + (size_t)b * SCH * T_ + t;
  float M = -__FLT_MAX__;
#pragma unroll
  for (int c = 0; c < SCH; ++c) M = fmaxf(M, pmp[(size_t)c * T_]);
  float Ssum = 0.0f;
#pragma unroll
  for (int c = 0; c < SCH; ++c)
    Ssum += psp[(size_t)c * T_] * __expf(pmp[(size_t)c * T_] - M);
  log_v[i] = log_nu[i] - (M + __logf(Ssum));
}

// ---------------------------------------------------------------------------
// final: out[0..n) = exp(log_pi), out[n..2n) = log_pi  (tuple order)
// ---------------------------------------------------------------------------
__global__ void sinkhorn_final(const float* __restrict__ cost,
                               const float* __restrict__ log_u,
                               const float* __restrict__ log_v,
                               float* __restrict__ out) {
  const size_t n = (size_t)B_ * S_ * T_;
  size_t i = ((size_t)blockIdx.x * blockDim.x + threadIdx.x) * 4;
  if (i >= n) return;
  size_t bs = i / T_;                  // T_ % 4 == 0 -> 4 elems share a row
  int b = (int)(bs / S_);
  int t = (int)(i % T_);

  float lu = log_u[bs];
  const float4 c  = *(const float4*)(cost + i);
  const float4 lv = *(const float4*)(log_v + (size_t)b * T_ + t);
  float4 lp;
  lp.x = lu + fmaf(-EPS_INV, c.x, lv.x);
  lp.y = lu + fmaf(-EPS_INV, c.y, lv.y);
  lp.z = lu + fmaf(-EPS_INV, c.z, lv.z);
  lp.w = lu + fmaf(-EPS_INV, c.w, lv.w);
  float4 tr;
  tr.x = __expf(lp.x);
  tr.y = __expf(lp.y);
  tr.z = __expf(lp.z);
  tr.w = __expf(lp.w);
  *(float4*)(out + i)     = tr;   // transport
  *(float4*)(out + n + i) = lp;   // log_pi
}

// ---------------------------------------------------------------------------
extern "C" void kernel_launch(void* const* d_in, const int* in_sizes, int n_in,
                              void* d_out, int out_size, void* d_ws, size_t ws_size,
                              hipStream_t stream) {
  const float* cost = (const float*)d_in[0];   // (B,S,T)
  const float* mu   = (const float*)d_in[1];   // (B,S)
  const float* nu   = (const float*)d_in[2];   // (B,T)

  float* ws     = (float*)d_ws;
  float* log_mu = ws;                                   // B*S
  float* log_nu = log_mu + (size_t)B_ * S_;             // B*T
  float* log_u  = log_nu + (size_t)B_ * T_;             // B*S
  float* log_v  = log_u  + (size_t)B_ * S_;             // B*T
  float* pm     = log_v  + (size_t)B_ * T_;             // B*SCH*T
  float* ps     = pm     + (size_t)B_ * SCH * T_;       // B*SCH*T

  sinkhorn_init<<<(B_ * S_ + 255) / 256, 256, 0, stream>>>(
      mu, nu, log_mu, log_nu, log_u, log_v);

  for (int it = 0; it < ITERS; ++it) {
    sinkhorn_row_update<<<B_ * (S_ / 16), 32 * RWAVES, 0, stream>>>(
        cost, log_mu, log_v, log_u);
    sinkhorn_col_partial<<<B_ * (T_ / TCOLS) * SCH, TCOLS, 0, stream>>>(
        cost, log_u, pm, ps);
    sinkhorn_col_combine<<<(B_ * T_ + 255) / 256, 256, 0, stream>>>(
        pm, ps, log_nu, log_v);
  }

  size_t nquads = ((size_t)B_ * S_ * T_) / 4;
  sinkhorn_final<<<(int)((nquads + 255) / 256), 256, 0, stream>>>(
      cost, log_u, log_v, (float*)d_out);
}